// QAModelV2_38534446579852
// MI455X (gfx1250) — compile-verified
//
#include <hip/hip_runtime.h>
#include <hip/hip_bf16.h>
#include <math.h>

typedef float v2f __attribute__((ext_vector_type(2)));
typedef float v8f __attribute__((ext_vector_type(8)));

#define B_  8
#define S_  4096
#define H_  1024
#define NS_ 64
#define NP_ 32
#define A_  8

// ---------------------------------------------------------------------------
// Kernel 1: P[token*4 + n] = seq[token,:] . W[:,n]   for n in {qa0,qa1,rank,sp}
// One wave (32 lanes) owns a 16-token tile. K-loop step 4 with fp32 WMMA.
// A (16x4 f32) layout: lanes 0-15 row M=lane  K={k0,k0+1}; lanes 16-31 row
// M=lane-16 K={k0+2,k0+3}.  B (4x16 f32): vgpr0 lanes0-15 K=k0 N=lane,
// lanes16-31 K=k0+1; vgpr1 K=k0+2 / k0+3.  D 16x16: vgpr v, lane l holds
// M = v + (l>=16 ? 8 : 0), N = l&15.  Only N<4 columns are real.
// ---------------------------------------------------------------------------
__global__ void proj_kernel(const float* __restrict__ seq,
                            const float* __restrict__ Wqa,
                            const float* __restrict__ Wrank,
                            const float* __restrict__ Wsp,
                            float* __restrict__ P) {
  __shared__ float lW[H_ * 4]; // combined weights, 16 KB
  for (int i = threadIdx.x; i < H_; i += blockDim.x) {
    lW[i * 4 + 0] = Wqa[i * 2 + 0];
    lW[i * 4 + 1] = Wqa[i * 2 + 1];
    lW[i * 4 + 2] = Wrank[i];
    lW[i * 4 + 3] = Wsp[i];
  }
  __syncthreads();

  const int lane   = threadIdx.x & 31;
  const int waveId = threadIdx.x >> 5;
  const int tile   = blockIdx.x * (blockDim.x >> 5) + waveId;
  const int token0 = tile * 16;
  if (token0 >= B_ * S_) return;   // wave-uniform

  const int m    = lane & 15;
  const int half = lane >> 4;                 // 0 or 1 (K sub-pair)
  const float* rowA = seq + (size_t)(token0 + m) * H_ + 2 * half;
  const int n  = lane & 15;                   // output column this lane feeds B for
  const int nn = (n < 4) ? n : 0;             // clamp so LDS reads stay in-bounds
  const float* seqEnd = seq + (size_t)B_ * S_ * H_;

  v8f c = {};
  for (int k0 = 0; k0 < H_; k0 += 32) {
    // one branch-free stream-prefetch ~2KB ahead per 8-WMMA group; the pointer
    // select (no branch) keeps the unrolled bodies in a single basic block so
    // the 8 global_load_b64 below can be clause-grouped with staggered waits.
    const float* pf = rowA + k0 + 512;
    __builtin_prefetch((pf < seqEnd) ? pf : seq, 0, 1);
    #pragma unroll
    for (int kk = 0; kk < 32; kk += 4) {
      const int k = k0 + kk;
      v2f a;
      a.x = rowA[k + 0];
      a.y = rowA[k + 1];
      float b0 = lW[(k + half) * 4 + nn];
      float b1 = lW[(k + 2 + half) * 4 + nn];
      v2f b;
      b.x = (n < 4) ? b0 : 0.0f;             // branch-free: EXEC stays all-ones
      b.y = (n < 4) ? b1 : 0.0f;
      c = __builtin_amdgcn_wmma_f32_16x16x4_f32(false, a, false, b,
                                                (short)0, c, false, false);
    }
  }

  if (n < 4) {                                // only 4 lanes per half write
    const int rowBase = token0 + (half ? 8 : 0);
    #pragma unroll
    for (int v = 0; v < 8; ++v)
      P[(size_t)(rowBase + v) * 4 + n] = c[v];
  }
}

// ---------------------------------------------------------------------------
// Block reduction helpers (256 threads)
// ---------------------------------------------------------------------------
__device__ float blk_sum(float v, float* red) {
  int t = threadIdx.x;
  red[t] = v; __syncthreads();
  for (int off = 128; off > 0; off >>= 1) {
    if (t < off) red[t] += red[t + off];
    __syncthreads();
  }
  float r = red[0]; __syncthreads();
  return r;
}
__device__ float blk_max(float v, float* red) {
  int t = threadIdx.x;
  red[t] = v; __syncthreads();
  for (int off = 128; off > 0; off >>= 1) {
    if (t < off) red[t] = fmaxf(red[t], red[t + off]);
    __syncthreads();
  }
  float r = red[0]; __syncthreads();
  return r;
}

// ---------------------------------------------------------------------------
// Kernel 2: per-batch reductions. One block (256 threads) per batch.
// partials[b*3+0]=rank, +1=sp, +2=span
// ---------------------------------------------------------------------------
__global__ void reduce_kernel(const float* __restrict__ P,
                              const int* __restrict__ tt,
                              const int* __restrict__ sent_starts, // (B, NS+1)
                              const int* __restrict__ para_starts, // (B, NP)
                              const float* __restrict__ para_labels,
                              const float* __restrict__ sp_labels, // (B, NS)
                              const int* __restrict__ starts,      // (B, A)
                              const int* __restrict__ ends,        // (B, A)
                              const float* __restrict__ bqa,
                              const float* __restrict__ brank,
                              const float* __restrict__ bsp,
                              float* __restrict__ partials) {
  __shared__ float red[256];
  __shared__ float fsR[NP_ + 1], flR[NP_ + 1];
  __shared__ float fsS[NS_ + 1], flS[NS_ + 1];

  const int b = blockIdx.x;
  const int t = threadIdx.x;
  const float NEG_INF = -__builtin_inff();
  const float bqa0 = bqa[0], bqa1 = bqa[1], br0 = brank[0], bs0 = bsp[0];
  const float* Pb = P + (size_t)b * S_ * 4;

  // ---- masked log-softmax normalizers for start / end logits ----
  float m0 = NEG_INF, m1 = NEG_INF;
  for (int s = t; s < S_; s += 256) {
    if (tt[b * S_ + s] == 1) {
      m0 = fmaxf(m0, Pb[s * 4 + 0] + bqa0);
      m1 = fmaxf(m1, Pb[s * 4 + 1] + bqa1);
    }
  }
  const float M0 = blk_max(m0, red);
  const float M1 = blk_max(m1, red);
  float e0 = 0.f, e1 = 0.f;
  for (int s = t; s < S_; s += 256) {
    if (tt[b * S_ + s] == 1) {
      e0 += expf(Pb[s * 4 + 0] + bqa0 - M0);
      e1 += expf(Pb[s * 4 + 1] + bqa1 - M1);
    }
  }
  const float logZ0 = M0 + logf(blk_sum(e0, red));
  const float logZ1 = M1 + logf(blk_sum(e1, red));

  // ---- sp sentence scores (segment means of the Wsp projection) ----
  if (t < NS_) {
    const int st = sent_starts[b * (NS_ + 1) + t];
    const int en = sent_starts[b * (NS_ + 1) + t + 1];
    float sc;
    if (en == -1) {
      sc = Pb[(S_ - 1) * 4 + 3] + bs0;           // seq[:, -1, :] . Wsp + bsp
    } else {
      int st_c = min(max(st, 0), S_);
      int en_c = max(min(max(en, 0), S_), st_c + 1);
      float acc = 0.f;
      for (int r = st_c; r < en_c && r < S_; ++r) acc += Pb[r * 4 + 3];
      sc = acc / (float)(en_c - st_c) + bs0;
    }
    fsS[t + 1] = sc;
    flS[t + 1] = sp_labels[b * NS_ + t];
  }
  // ---- para scores ----
  if (t < NP_) {
    const int ps = para_starts[b * NP_ + t];
    fsR[t + 1] = Pb[ps * 4 + 2] + br0;
    flR[t + 1] = para_labels[b * NP_ + t];
  }
  if (t == 0) {
    fsR[0] = Pb[2] + br0;  flR[0] = 0.5f;        // rank threshold (token 0)
    fsS[0] = Pb[3] + bs0;  flS[0] = 0.5f;        // sp threshold (token 0)
  }
  __syncthreads();

  // ---- listMLE (rank, 33 entries) ----
  float lpR = 0.f;
  if (t < NP_ + 1) {
    const float fj = fsR[t], lj = flR[t];
    float rest = 0.f;
    for (int k = 0; k < NP_ + 1; ++k) {
      const float sk = expf(fsR[k]) * ((flR[k] != -1.f) ? 1.f : 0.f);
      rest += (flR[k] < lj) ? sk : 0.f;
    }
    if (lj >= 0.5f) {
      const float sj = expf(fj) * ((lj != -1.f) ? 1.f : 0.f);
      lpR = fj - logf(sj + rest);
    }
  }
  const float rank_partial = -blk_sum(lpR, red);

  // ---- listMLE (sp, 65 entries) ----
  float lpS = 0.f;
  if (t < NS_ + 1) {
    const float fj = fsS[t], lj = flS[t];
    float rest = 0.f;
    for (int k = 0; k < NS_ + 1; ++k) {
      const float sk = expf(fsS[k]) * ((flS[k] != -1.f) ? 1.f : 0.f);
      rest += (flS[k] < lj) ? sk : 0.f;
    }
    if (lj >= 0.5f) {
      const float sj = expf(fj) * ((lj != -1.f) ? 1.f : 0.f);
      lpS = fj - logf(sj + rest);
    }
  }
  const float sp_partial = -blk_sum(lpS, red);

  // ---- span loss over A answers ----
  float ev = 0.f;
  if (t < A_) {
    const int si = starts[b * A_ + t];
    const int ei = ends[b * A_ + t];
    float ce = 0.f;
    if (si != -1) {
      const int ss = min(max(si, 0), S_ - 1);
      const float l = (tt[b * S_ + ss] == 1) ? (Pb[ss * 4 + 0] + bqa0) : NEG_INF;
      ce += -(l - logZ0);
    }
    if (ei != -1) {
      const int se = min(max(ei, 0), S_ - 1);
      const float l = (tt[b * S_ + se] == 1) ? (Pb[se * 4 + 1] + bqa1) : NEG_INF;
      ce += -(l - logZ1);
    }
    float lp = -ce;
    if (lp == 0.0f) lp = NEG_INF;                // matches jnp.where(lp == 0.0, -inf, lp)
    ev = expf(lp);
  }
  const float msum = blk_sum(ev, red);
  const float span_partial = (msum > 0.f) ? -logf(msum) : 0.f;

  if (t == 0) {
    partials[b * 3 + 0] = rank_partial;
    partials[b * 3 + 1] = sp_partial;
    partials[b * 3 + 2] = span_partial;
  }
}

// ---------------------------------------------------------------------------
// Kernel 3: fold 8 batch partials -> (loss, rank, sp)
// ---------------------------------------------------------------------------
__global__ void final_kernel(const float* __restrict__ partials,
                             float* __restrict__ out) {
  if (threadIdx.x == 0) {
    float rank = 0.f, sp = 0.f, span = 0.f;
    for (int b = 0; b < B_; ++b) {
      rank += partials[b * 3 + 0];
      sp   += partials[b * 3 + 1];
      span += partials[b * 3 + 2];
    }
    out[0] = rank + span + sp;   // RANK_W = SP_W = ANS_W = 1.0
    out[1] = rank;
    out[2] = sp;
  }
}

extern "C" void kernel_launch(void* const* d_in, const int* in_sizes, int n_in,
                              void* d_out, int out_size, void* d_ws, size_t ws_size,
                              hipStream_t stream) {
  const float* seq         = (const float*)d_in[0];
  const int*   tt          = (const int*)  d_in[1];
  const int*   sent_starts = (const int*)  d_in[2];
  const int*   para_starts = (const int*)  d_in[3];
  const float* para_labels = (const float*)d_in[4];
  const float* sp_labels   = (const float*)d_in[5];
  const int*   starts      = (const int*)  d_in[6];
  const int*   ends        = (const int*)  d_in[7];
  const float* Wqa         = (const float*)d_in[8];
  const float* bqa         = (const float*)d_in[9];
  const float* Wrank       = (const float*)d_in[10];
  const float* brank       = (const float*)d_in[11];
  const float* Wsp         = (const float*)d_in[12];
  const float* bsp         = (const float*)d_in[13];

  float* P        = (float*)d_ws;              // B*S*4 floats = 512 KB
  float* partials = P + (size_t)B_ * S_ * 4;   // B*3 floats

  const int tiles = (B_ * S_) / 16;            // 2048 tiles, 8 waves/block
  proj_kernel<<<tiles / 8, 256, 0, stream>>>(seq, Wqa, Wrank, Wsp, P);
  reduce_kernel<<<B_, 256, 0, stream>>>(P, tt, sent_starts, para_starts,
                                        para_labels, sp_labels, starts, ends,
                                        bqa, brank, bsp, partials);
  final_kernel<<<1, 32, 0, stream>>>(partials, (float*)d_out);
}